// EncoderLayer_70600672411681
// MI455X (gfx1250) — compile-verified
//
#include <hip/hip_runtime.h>
#include <math.h>

#define EPSF 1e-5f

typedef float v2f __attribute__((ext_vector_type(2)));
typedef float v8f __attribute__((ext_vector_type(8)));

#if __has_builtin(__builtin_amdgcn_wmma_f32_16x16x4_f32)
#define HAVE_WMMA_F32 1
#else
#define HAVE_WMMA_F32 0
#endif

struct SelState {
  unsigned prefix, need, greater, threshold, totalG, needTies;
};

// ---------------- utility fills ----------------
__global__ void fill_f32_kernel(float* p, float v, long long n) {
  long long i = (long long)blockIdx.x * blockDim.x + threadIdx.x;
  if (i < n) p[i] = v;
}
__global__ void fill_i32_kernel(int* p, int v, long long n) {
  long long i = (long long)blockIdx.x * blockDim.x + threadIdx.x;
  if (i < n) p[i] = v;
}

// ---------------- pooling: score = tanh(x.w / ||w||) ----------------
__global__ void norm_kernel(const float* __restrict__ pw, float* __restrict__ wn) {
  __shared__ float sm[128];
  float v = pw[threadIdx.x];
  sm[threadIdx.x] = v * v;
  __syncthreads();
  for (int s = 64; s; s >>= 1) {
    if ((int)threadIdx.x < s) sm[threadIdx.x] += sm[threadIdx.x + s];
    __syncthreads();
  }
  if (threadIdx.x == 0) wn[0] = 1.0f / sqrtf(sm[0]);
}

__global__ __launch_bounds__(256) void score_kernel(
    const float* __restrict__ x, const float* __restrict__ pw,
    const float* __restrict__ wn, float* __restrict__ scores,
    unsigned* __restrict__ keys, int N) {
  int gid = blockIdx.x * blockDim.x + threadIdx.x;
  int node = gid >> 5;
  int lane = threadIdx.x & 31;
  if (node >= N) return;
  const float4 xv = *(const float4*)&x[(long long)node * 128 + lane * 4];
  const float4 wv = *(const float4*)&pw[lane * 4];
  float s = xv.x * wv.x + xv.y * wv.y + xv.z * wv.z + xv.w * wv.w;
  for (int off = 16; off; off >>= 1) s += __shfl_xor(s, off, 32);
  if (lane == 0) {
    float sc = tanhf(s * wn[0]);
    scores[node] = sc;
    unsigned u = __float_as_uint(sc);
    keys[node] = (u & 0x80000000u) ? ~u : (u | 0x80000000u);  // descending-orderable
  }
}

// ---------------- exact top-K threshold via 4x8-bit radix select ----------------
__global__ void initsel_kernel(SelState* st, unsigned* hist, int K) {
  if (threadIdx.x == 0) {
    st->prefix = 0u; st->need = (unsigned)K; st->greater = 0u;
    st->threshold = 0u; st->totalG = 0u; st->needTies = 0u;
  }
  hist[threadIdx.x] = 0u;
}

__global__ __launch_bounds__(256) void hist_kernel(
    const unsigned* __restrict__ keys, int N, int pass,
    const SelState* __restrict__ st, unsigned* __restrict__ hist) {
  int i = blockIdx.x * blockDim.x + threadIdx.x;
  if (i >= N) return;
  unsigned key = keys[i];
  unsigned mask = pass ? (0xFFFFFFFFu << (32 - 8 * pass)) : 0u;
  if ((key & mask) == st->prefix) {
    int shift = 24 - 8 * pass;
    atomicAdd(&hist[(key >> shift) & 255u], 1u);
  }
}

__global__ void select_kernel(SelState* st, unsigned* hist, int pass) {
  __shared__ unsigned h[256];
  h[threadIdx.x] = hist[threadIdx.x];
  __syncthreads();
  if (threadIdx.x == 0) {
    unsigned need = st->need, greater = st->greater, prefix = st->prefix;
    int shift = 24 - 8 * pass;
    for (int b = 255; b >= 0; --b) {
      unsigned cnt = h[b];
      if (need > cnt) { need -= cnt; greater += cnt; }
      else { prefix |= ((unsigned)b) << shift; break; }
    }
    st->need = need; st->greater = greater; st->prefix = prefix;
    if (pass == 3) { st->threshold = prefix; st->totalG = greater; st->needTies = need; }
  }
  __syncthreads();
  hist[threadIdx.x] = 0u;  // ready for next pass
}

// ---------------- deterministic compaction: perm + gate values ----------------
__global__ __launch_bounds__(256) void count_kernel(
    const unsigned* __restrict__ keys, int N, const SelState* __restrict__ st,
    unsigned* __restrict__ blockG, unsigned* __restrict__ blockE) {
  __shared__ unsigned sg[256], se[256];
  int t = threadIdx.x;
  int i = blockIdx.x * 256 + t;
  unsigned T = st->threshold;
  unsigned g = 0, e = 0;
  if (i < N) { unsigned k = keys[i]; g = (k > T) ? 1u : 0u; e = (k == T) ? 1u : 0u; }
  sg[t] = g; se[t] = e;
  __syncthreads();
  for (int s = 128; s; s >>= 1) {
    if (t < s) { sg[t] += sg[t + s]; se[t] += se[t + s]; }
    __syncthreads();
  }
  if (t == 0) { blockG[blockIdx.x] = sg[0]; blockE[blockIdx.x] = se[0]; }
}

__global__ void scan_kernel(const unsigned* bG, const unsigned* bE,
                            unsigned* oG, unsigned* oE, int nb) {
  unsigned aG = 0, aE = 0;
  for (int i = 0; i < nb; ++i) {
    oG[i] = aG; aG += bG[i];
    oE[i] = aE; aE += bE[i];
  }
}

__global__ __launch_bounds__(256) void scatter_kernel(
    const unsigned* __restrict__ keys, const float* __restrict__ scores, int N,
    const SelState* __restrict__ st, const unsigned* __restrict__ oG,
    const unsigned* __restrict__ oE, int* __restrict__ permI, float* __restrict__ topv) {
  __shared__ unsigned sg[256], se[256];
  int t = threadIdx.x;
  int i = blockIdx.x * 256 + t;
  unsigned T = st->threshold;
  unsigned g = 0, e = 0;
  if (i < N) { unsigned k = keys[i]; g = (k > T) ? 1u : 0u; e = (k == T) ? 1u : 0u; }
  sg[t] = g; se[t] = e;
  __syncthreads();
  for (int s = 1; s < 256; s <<= 1) {  // inclusive Hillis-Steele scan
    unsigned vg = 0, ve = 0;
    if (t >= s) { vg = sg[t - s]; ve = se[t - s]; }
    __syncthreads();
    sg[t] += vg; se[t] += ve;
    __syncthreads();
  }
  if (i < N) {
    if (g) {
      unsigned slot = oG[blockIdx.x] + sg[t] - 1u;
      permI[slot] = i; topv[slot] = scores[i];
    } else if (e) {
      unsigned rank = oE[blockIdx.x] + se[t] - 1u;
      if (rank < st->needTies) {
        unsigned slot = st->totalG + rank;
        permI[slot] = i; topv[slot] = scores[i];
      }
    }
  }
}

__global__ void mapset_kernel(const int* __restrict__ permI, int K, int* __restrict__ mapping) {
  int k = blockIdx.x * blockDim.x + threadIdx.x;
  if (k < K) mapping[permI[k]] = k;
}

__global__ __launch_bounds__(256) void gather_kernel(
    const float* __restrict__ x, const int* __restrict__ permI,
    const float* __restrict__ topv, float* __restrict__ xp, int K) {
  int gid = blockIdx.x * blockDim.x + threadIdx.x;
  int k = gid >> 5;
  int lane = threadIdx.x & 31;
  if (k >= K) return;
  int src = permI[k];
  float g = topv[k];
  float4 v = *(const float4*)&x[(long long)src * 128 + lane * 4];
  v.x *= g; v.y *= g; v.z *= g; v.w *= g;
  *(float4*)&xp[(long long)k * 128 + lane * 4] = v;
}

// ---------------- edge remap ----------------
__global__ __launch_bounds__(256) void edge_kernel(
    const int* __restrict__ ei, const float* __restrict__ ew,
    const int* __restrict__ mapping, int E,
    int* __restrict__ rowA, int* __restrict__ colA,
    float* __restrict__ eiOut, float* __restrict__ ewOut) {
  int e = blockIdx.x * blockDim.x + threadIdx.x;
  if (e >= E) return;
  int r0 = mapping[ei[e]];
  int c0 = mapping[ei[(long long)E + e]];
  bool valid = (r0 >= 0) && (c0 >= 0);
  int r = valid ? r0 : 0;
  int c = valid ? c0 : 0;
  float w = valid ? ew[e] : 0.0f;
  rowA[e] = r; colA[e] = c;
  eiOut[e] = (float)r; eiOut[(long long)E + e] = (float)c;
  ewOut[e] = w;
}

// ---------------- fp32 WMMA GEMM: Out[M,128] = A[M,128] @ W[128,128]^T ----------------
__global__ __launch_bounds__(256) void gemm128_kernel(
    const float* __restrict__ A, const float* __restrict__ W,
    const float* __restrict__ bias, float* __restrict__ Out,
    int M, int applyRelu) {
  __shared__ float As[16][132];  // padded to dodge bank conflicts
  const int t = threadIdx.x;
  const int rowBase = blockIdx.x * 16;
  // cooperative load of a 16x128 A strip into LDS (zero-padded past M)
  for (int i = 0; i < 2; ++i) {
    int idx = t + i * 256;               // 0..511 float4 slots
    int r = idx >> 5, c4 = (idx & 31) << 2;
    int gr = rowBase + r;
    float4 a4 = make_float4(0.f, 0.f, 0.f, 0.f);
    if (gr < M) a4 = *(const float4*)&A[(long long)gr * 128 + c4];
    As[r][c4 + 0] = a4.x; As[r][c4 + 1] = a4.y;
    As[r][c4 + 2] = a4.z; As[r][c4 + 3] = a4.w;
  }
  __syncthreads();
  const int lane = t & 31, wv = t >> 5;
  const int half = lane >> 4, ml = lane & 15;
  const int ncol = wv * 16 + ml;          // wave's 16-column tile
  v8f c = {0.f, 0.f, 0.f, 0.f, 0.f, 0.f, 0.f, 0.f};
#if HAVE_WMMA_F32
  for (int kk = 0; kk < 128; kk += 4) {
    // A 16x4 frag: lanes 0-15 hold K=kk,kk+1; lanes 16-31 hold K=kk+2,kk+3
    v2f a = *(const v2f*)&As[ml][kk + 2 * half];
    // B 4x16 frag: B[k][n] = W[n][k]; same lane->k pattern, n = ncol
    v2f b = *(const v2f*)&W[(long long)ncol * 128 + kk + 2 * half];
    c = __builtin_amdgcn_wmma_f32_16x16x4_f32(false, a, false, b, (short)0, c,
                                              false, false);
  }
#else
  for (int r = 0; r < 8; ++r) {
    float acc = 0.f;
    int m = r + 8 * half;
    for (int k2 = 0; k2 < 128; ++k2) acc += As[m][k2] * W[(long long)ncol * 128 + k2];
    c[r] = acc;
  }
#endif
  // C/D layout: VGPR r -> row r (lanes 0-15) / row r+8 (lanes 16-31), col = ncol
  for (int r = 0; r < 8; ++r) {
    int grow = rowBase + r + 8 * half;
    if (grow < M) {
      float v = c[r] + (bias ? bias[ncol] : 0.f);
      if (applyRelu) v = fmaxf(v, 0.f);
      Out[(long long)grow * 128 + ncol] = v;
    }
  }
}

// ---------------- batchnorm (training-mode, biased variance) ----------------
__global__ __launch_bounds__(128) void bnstat_kernel(
    const float* __restrict__ h, int rows, float* __restrict__ sum, float* __restrict__ sq) {
  int c = threadIdx.x;
  int r0 = blockIdx.x * 256;
  int r1 = r0 + 256; if (r1 > rows) r1 = rows;
  float s = 0.f, q = 0.f;
  for (int r = r0; r < r1; ++r) {
    float v = h[(long long)r * 128 + c];
    s += v; q += v * v;
  }
  atomicAdd(&sum[c], s);
  atomicAdd(&sq[c], q);
}

__global__ void bnfin_kernel(const float* sum, const float* sq,
                             const float* g, const float* b, int rows,
                             float* scale, float* shift) {
  int c = threadIdx.x;
  float inv = 1.0f / (float)rows;
  float m = sum[c] * inv;
  float v = sq[c] * inv - m * m;
  if (v < 0.f) v = 0.f;
  float sc = g[c] * rsqrtf(v + EPSF);
  scale[c] = sc;
  shift[c] = b[c] - m * sc;
}

__global__ __launch_bounds__(256) void bn_apply_kernel(
    float* __restrict__ h, const float* __restrict__ scale,
    const float* __restrict__ shift, long long n) {
  long long i = (long long)blockIdx.x * blockDim.x + threadIdx.x;
  if (i >= n) return;
  int c = (int)(i & 127);
  h[i] = h[i] * scale[c] + shift[c];
}

__global__ __launch_bounds__(256) void bn_apply_relu2_kernel(
    float* __restrict__ r0, float* __restrict__ r5,
    const float* __restrict__ scale, const float* __restrict__ shift, long long n) {
  long long i = (long long)blockIdx.x * blockDim.x + threadIdx.x;
  if (i >= n) return;
  int c = (int)(i & 127);
  float v = r0[i] * scale[c] + shift[c];
  v = fmaxf(v, 0.f);
  r0[i] = v; r5[i] = v;
}

// ---------------- GCN degree / normalization / aggregation ----------------
__global__ __launch_bounds__(256) void deg_kernel(
    const int* __restrict__ colA, const float* __restrict__ ewOut, int E,
    float* __restrict__ deg) {
  int e = blockIdx.x * blockDim.x + threadIdx.x;
  if (e >= E) return;
  float w = ewOut[e];
  if (w != 0.f) atomicAdd(&deg[colA[e]], w);
}

__global__ __launch_bounds__(256) void dis_kernel(const float* __restrict__ deg,
                                                  float* __restrict__ dis, int K) {
  int k = blockIdx.x * blockDim.x + threadIdx.x;
  if (k >= K) return;
  float d = deg[k] + 2.0f;  // improved self-loop weight folded in (always > 0)
  dis[k] = rsqrtf(fmaxf(d, EPSF));
}

// 2 edges per 256-thread block; 128 channels each; L2-resident scatter-add
__global__ __launch_bounds__(256) void agg_kernel(
    const int* __restrict__ rowA, const int* __restrict__ colA,
    const float* __restrict__ ewOut, const float* __restrict__ dis,
    const float* __restrict__ h2, float* __restrict__ out0, int E) {
  int e = blockIdx.x * 2 + (threadIdx.x >> 7);
  int c = threadIdx.x & 127;
  if (e >= E) return;
  float w = ewOut[e];
  if (w == 0.f) return;
  int r = rowA[e], cl = colA[e];
  float nrm = dis[r] * w * dis[cl];
  atomicAdd(&out0[(long long)cl * 128 + c], h2[(long long)r * 128 + c] * nrm);
}

__global__ __launch_bounds__(256) void self_kernel(
    const float* __restrict__ h2, const float* __restrict__ dis,
    const float* __restrict__ gb, float* __restrict__ out0, int K) {
  long long i = (long long)blockIdx.x * blockDim.x + threadIdx.x;
  long long n = (long long)K * 128;
  if (i >= n) return;
  int k = (int)(i >> 7), c = (int)(i & 127);
  float d = dis[k];
  out0[i] += h2[i] * (2.0f * d * d) + gb[c];
}

// ---------------- tail outputs: batch_p, perm ----------------
__global__ void tail_kernel(const int* __restrict__ permI, const int* __restrict__ batch,
                            float* __restrict__ r3, float* __restrict__ r4, int K) {
  int k = blockIdx.x * blockDim.x + threadIdx.x;
  if (k >= K) return;
  int p = permI[k];
  r3[k] = (float)batch[p];
  r4[k] = (float)p;
}

// ---------------- host launcher ----------------
extern "C" void kernel_launch(void* const* d_in, const int* in_sizes, int n_in,
                              void* d_out, int out_size, void* d_ws, size_t ws_size,
                              hipStream_t stream) {
  const float* x      = (const float*)d_in[0];
  const int*   ei     = (const int*)d_in[1];
  const float* ew     = (const float*)d_in[2];
  const int*   batch  = (const int*)d_in[3];
  const float* pool_w = (const float*)d_in[4];
  const float* lin_w  = (const float*)d_in[5];
  const float* lin_b  = (const float*)d_in[6];
  const float* bn1_g  = (const float*)d_in[7];
  const float* bn1_b  = (const float*)d_in[8];
  const float* gcn_w  = (const float*)d_in[9];
  const float* gcn_b  = (const float*)d_in[10];
  const float* bn2_g  = (const float*)d_in[11];
  const float* bn2_b  = (const float*)d_in[12];
  (void)n_in; (void)out_size; (void)ws_size;

  const int C = in_sizes[4];             // 128 (kernels assume this)
  const int N = in_sizes[0] / C;
  const int E = in_sizes[2];
  const int K = (N + 1) / 2;
  const int nb = (N + 255) / 256;

  // workspace carve-out
  char* wp = (char*)d_ws;
  auto carve = [&](size_t bytes) -> void* {
    void* p = (void*)wp;
    wp += (bytes + 255) & ~(size_t)255;
    return p;
  };
  float*    scores  = (float*)carve((size_t)N * 4);
  unsigned* keys    = (unsigned*)carve((size_t)N * 4);
  int*      mapping = (int*)carve((size_t)N * 4);
  int*      permI   = (int*)carve((size_t)K * 4);
  float*    topv    = (float*)carve((size_t)K * 4);
  float*    xp      = (float*)carve((size_t)K * 128 * 4);
  float*    h1      = (float*)carve((size_t)K * 128 * 4);
  float*    h2      = (float*)carve((size_t)K * 128 * 4);
  float*    deg     = (float*)carve((size_t)K * 4);
  float*    dis     = (float*)carve((size_t)K * 4);
  int*      rowA    = (int*)carve((size_t)E * 4);
  int*      colA    = (int*)carve((size_t)E * 4);
  unsigned* hist    = (unsigned*)carve(256 * 4);
  SelState* st      = (SelState*)carve(sizeof(SelState));
  unsigned* blockG  = (unsigned*)carve((size_t)nb * 4);
  unsigned* blockE  = (unsigned*)carve((size_t)nb * 4);
  unsigned* offsG   = (unsigned*)carve((size_t)nb * 4);
  unsigned* offsE   = (unsigned*)carve((size_t)nb * 4);
  float*    sum1    = (float*)carve(128 * 4);
  float*    sq1     = (float*)carve(128 * 4);
  float*    sum2    = (float*)carve(128 * 4);
  float*    sq2     = (float*)carve(128 * 4);
  float*    scale1  = (float*)carve(128 * 4);
  float*    shift1  = (float*)carve(128 * 4);
  float*    scale2  = (float*)carve(128 * 4);
  float*    shift2  = (float*)carve(128 * 4);
  float*    wn      = (float*)carve(4);

  // output regions (tuple flattened in return order)
  float* r0 = (float*)d_out;                 // out       [K,128]
  float* r1 = r0 + (size_t)K * 128;          // edge idx  [2,E]
  float* r2 = r1 + 2 * (size_t)E;            // ew        [E]
  float* r3 = r2 + (size_t)E;                // batch_p   [K]
  float* r4 = r3 + (size_t)K;                // perm      [K]
  float* r5 = r4 + (size_t)K;                // out again [K,128]

  auto cdiv = [](long long a, long long b) -> unsigned { return (unsigned)((a + b - 1) / b); };
  const long long KC = (long long)K * 128;

  // 1) pooling scores + top-K selection
  norm_kernel<<<1, 128, 0, stream>>>(pool_w, wn);
  score_kernel<<<cdiv(N, 8), 256, 0, stream>>>(x, pool_w, wn, scores, keys, N);
  initsel_kernel<<<1, 256, 0, stream>>>(st, hist, K);
  for (int p = 0; p < 4; ++p) {
    hist_kernel<<<cdiv(N, 256), 256, 0, stream>>>(keys, N, p, st, hist);
    select_kernel<<<1, 256, 0, stream>>>(st, hist, p);
  }
  count_kernel<<<nb, 256, 0, stream>>>(keys, N, st, blockG, blockE);
  scan_kernel<<<1, 1, 0, stream>>>(blockG, blockE, offsG, offsE, nb);
  scatter_kernel<<<nb, 256, 0, stream>>>(keys, scores, N, st, offsG, offsE, permI, topv);

  // 2) node remap + gated gather
  fill_i32_kernel<<<cdiv(N, 256), 256, 0, stream>>>(mapping, -1, N);
  mapset_kernel<<<cdiv(K, 256), 256, 0, stream>>>(permI, K, mapping);
  gather_kernel<<<cdiv(K, 8), 256, 0, stream>>>(x, permI, topv, xp, K);
  edge_kernel<<<cdiv(E, 256), 256, 0, stream>>>(ei, ew, mapping, E, rowA, colA, r1, r2);

  // 3) MLP: Linear + ReLU (WMMA fp32), then BatchNorm1
  gemm128_kernel<<<cdiv(K, 16), 256, 0, stream>>>(xp, lin_w, lin_b, h1, K, 1);
  fill_f32_kernel<<<1, 128, 0, stream>>>(sum1, 0.f, 128);
  fill_f32_kernel<<<1, 128, 0, stream>>>(sq1, 0.f, 128);
  bnstat_kernel<<<cdiv(K, 256), 128, 0, stream>>>(h1, K, sum1, sq1);
  bnfin_kernel<<<1, 128, 0, stream>>>(sum1, sq1, bn1_g, bn1_b, K, scale1, shift1);
  bn_apply_kernel<<<cdiv(KC, 256), 256, 0, stream>>>(h1, scale1, shift1, KC);

  // 4) GCN: GEMM (WMMA fp32) + normalized aggregate
  gemm128_kernel<<<cdiv(K, 16), 256, 0, stream>>>(h1, gcn_w, nullptr, h2, K, 0);
  fill_f32_kernel<<<cdiv(K, 256), 256, 0, stream>>>(deg, 0.f, K);
  deg_kernel<<<cdiv(E, 256), 256, 0, stream>>>(colA, r2, E, deg);
  dis_kernel<<<cdiv(K, 256), 256, 0, stream>>>(deg, dis, K);
  fill_f32_kernel<<<cdiv(KC, 256), 256, 0, stream>>>(r0, 0.f, KC);
  agg_kernel<<<cdiv(E, 2), 256, 0, stream>>>(rowA, colA, r2, dis, h2, r0, E);
  self_kernel<<<cdiv(KC, 256), 256, 0, stream>>>(h2, dis, gcn_b, r0, K);

  // 5) BatchNorm2 + ReLU -> out (written to both duplicate output slots)
  fill_f32_kernel<<<1, 128, 0, stream>>>(sum2, 0.f, 128);
  fill_f32_kernel<<<1, 128, 0, stream>>>(sq2, 0.f, 128);
  bnstat_kernel<<<cdiv(K, 256), 128, 0, stream>>>(r0, K, sum2, sq2);
  bnfin_kernel<<<1, 128, 0, stream>>>(sum2, sq2, bn2_g, bn2_b, K, scale2, shift2);
  bn_apply_relu2_kernel<<<cdiv(KC, 256), 256, 0, stream>>>(r0, r5, scale2, shift2, KC);

  // 6) batch_p + perm outputs
  tail_kernel<<<cdiv(K, 256), 256, 0, stream>>>(permI, batch, r3, r4, K);
}